// FC_63960652972583
// MI455X (gfx1250) — compile-verified
//
#include <hip/hip_runtime.h>

typedef __attribute__((ext_vector_type(2))) float v2f;
typedef __attribute__((ext_vector_type(4))) float v4f;
typedef __attribute__((ext_vector_type(8))) float v8f;

namespace {
constexpr int kL   = 3;
constexpr int kN   = 120000;
constexpr int kD   = 16;
constexpr int kSel = 60000;
}  // namespace

// One wave32 per selected point j. The wave runs all 6 (block, layer) slices:
//   y = W(16x16) * f(16) + hb           -> 4x V_WMMA_F32_16X16X4_F32 (K=4 chain)
//   h[l,j,:] = y                        -> lane0 / lane16 store the two halves
//   out[j]  = (1/6) * sum_{b,l} (<y,tw> + tb)
__global__ __launch_bounds__(256) void fc_wmma_kernel(
    const float* __restrict__ feat0, const float* __restrict__ hw0,
    const float* __restrict__ hb0,   const float* __restrict__ tw0,
    const float* __restrict__ tb0,
    const float* __restrict__ feat1, const float* __restrict__ hw1,
    const float* __restrict__ hb1,   const float* __restrict__ tw1,
    const float* __restrict__ tb1,
    const int* __restrict__ mask,
    float* __restrict__ dout) {
  const int lane = (int)(threadIdx.x & 31u);
  const int m    = lane & 15;   // row within half
  const int hi   = lane >> 4;   // 0: lanes 0-15, 1: lanes 16-31

  int j = (int)blockIdx.x * 8 + (int)(threadIdx.x >> 5);
  j = __builtin_amdgcn_readfirstlane(j);   // wave-uniform selection index
  if (j >= kSel) return;

  const int n = __builtin_amdgcn_readfirstlane(mask[j]);  // gathered point id

  float* const outv = dout;                       // [kSel]
  float* const h0   = dout + kSel;                // [kL][kSel][kD]
  float* const h1   = h0 + (long)kL * kSel * kD;  // [kL][kSel][kD]

  float osum = 0.0f;

#pragma unroll
  for (int b = 0; b < 2; ++b) {
    const float* __restrict__ feat = b ? feat1 : feat0;
    const float* __restrict__ hw   = b ? hw1   : hw0;
    const float* __restrict__ hb   = b ? hb1   : hb0;
    const float* __restrict__ tw   = b ? tw1   : tw0;
    const float* __restrict__ tb   = b ? tb1   : tb0;
    float* __restrict__ hout = b ? h1 : h0;

#pragma unroll
    for (int l = 0; l < kL; ++l) {
      const long base = (long)l * kN + n;
      const float* __restrict__ W  = hw   + base * (kD * kD);  // row-major 16x16
      const float* __restrict__ F  = feat + base * kD;
      const float* __restrict__ Bp = hb   + base * kD;
      const float* __restrict__ T  = tw   + base * kD;
      const float tbv = tb[base];

      // Preload accumulator with hb per the 16x16 f32 C/D layout:
      // VGPR v, lanes 0-15 -> row v, lanes 16-31 -> row v+8 (col = lane&15).
      const v4f bA = *(const v4f*)(Bp + 8 * hi);
      const v4f bB = *(const v4f*)(Bp + 8 * hi + 4);
      v8f acc;
      acc[0] = bA.x; acc[1] = bA.y; acc[2] = bA.z; acc[3] = bA.w;
      acc[4] = bB.x; acc[5] = bB.y; acc[6] = bB.z; acc[7] = bB.w;

      // 16x16 matvec as 4 chained K=4 WMMAs.
      // A (16x4 f32): VGPR0 = {K0 | K2}, VGPR1 = {K1 | K3}
      //   -> lane (m,hi) holds W[m][4i+2hi], W[m][4i+2hi+1] (b64 load).
      // B (4x16 f32): column-replicated f chunk, same half/K striping.
#pragma unroll
      for (int i = 0; i < 4; ++i) {
        const v2f a  = *(const v2f*)(W + m * kD + i * 4 + hi * 2);
        const v2f bv = *(const v2f*)(F + i * 4 + hi * 2);
        acc = __builtin_amdgcn_wmma_f32_16x16x4_f32(
            false, a, false, bv, (short)0, acc, false, false);
      }
      // acc now holds (W*f + hb) replicated across the 16 columns.

      // h output: lane 0 stores rows 0-7, lane 16 stores rows 8-15.
      if (m == 0) {
        float* p = hout + ((long)l * kSel + j) * kD + hi * 8;
        v4f lo, hi4;
        lo.x  = acc[0]; lo.y  = acc[1]; lo.z  = acc[2]; lo.w  = acc[3];
        hi4.x = acc[4]; hi4.y = acc[5]; hi4.z = acc[6]; hi4.w = acc[7];
        *(v4f*)(p + 0) = lo;
        *(v4f*)(p + 4) = hi4;
      }

      // Half-dot with tw, then fold the two halves across lanes.
      const v4f tA = *(const v4f*)(T + 8 * hi);
      const v4f tB = *(const v4f*)(T + 8 * hi + 4);
      float part = acc[0] * tA.x + acc[1] * tA.y + acc[2] * tA.z + acc[3] * tA.w +
                   acc[4] * tB.x + acc[5] * tB.y + acc[6] * tB.z + acc[7] * tB.w;
      part += __shfl_xor(part, 16, 32);
      osum += part + tbv;
    }
  }

  if (lane == 0) outv[j] = osum * (1.0f / 6.0f);
}

extern "C" void kernel_launch(void* const* d_in, const int* in_sizes, int n_in,
                              void* d_out, int out_size, void* d_ws, size_t ws_size,
                              hipStream_t stream) {
  (void)in_sizes; (void)n_in; (void)out_size; (void)d_ws; (void)ws_size;
  const float* feat0 = (const float*)d_in[0];
  const float* hw0   = (const float*)d_in[1];
  const float* hb0   = (const float*)d_in[2];
  const float* tw0   = (const float*)d_in[3];
  const float* tb0   = (const float*)d_in[4];
  const float* feat1 = (const float*)d_in[5];
  const float* hw1   = (const float*)d_in[6];
  const float* hb1   = (const float*)d_in[7];
  const float* tw1   = (const float*)d_in[8];
  const float* tb1   = (const float*)d_in[9];
  const int*   mask  = (const int*)d_in[10];
  float* out = (float*)d_out;

  const int waves_per_block = 8;  // 256 threads = 8 wave32
  const int blocks = (kSel + waves_per_block - 1) / waves_per_block;  // 7500
  fc_wmma_kernel<<<blocks, 256, 0, stream>>>(feat0, hw0, hb0, tw0, tb0,
                                             feat1, hw1, hb1, tw1, tb1,
                                             mask, out);
}